// CRF_33389075759541
// MI455X (gfx1250) — compile-verified
//
#include <hip/hip_runtime.h>
#include <hip/hip_bf16.h>

#define NUM_TAGS 64
#define SEQ_LEN  2048
#define BSZ      256
#define BPG      16     // batches per workgroup (WMMA M dimension)
#define LSTRIDE  68     // LDS row stride in floats (bank-conflict padding)

#define LOG2E 1.4426950408889634f
#define LN2   0.6931471805599453f

typedef float v2f __attribute__((ext_vector_type(2)));
typedef float v8f __attribute__((ext_vector_type(8)));

// max-reduce over 8 consecutive lanes (xor 1,3,4) using v_permlane16_b32
__device__ __forceinline__ float permlane_max8(float v)
{
    union { float f; unsigned u; } a, b;
    a.f = v;
    b.u = __builtin_amdgcn_permlane16(a.u, a.u, 0x67452301u, 0xEFCDAB89u, false, false);
    a.f = fmaxf(a.f, b.f);
    b.u = __builtin_amdgcn_permlane16(a.u, a.u, 0x45670123u, 0xCDEF89ABu, false, false);
    a.f = fmaxf(a.f, b.f);
    b.u = __builtin_amdgcn_permlane16(a.u, a.u, 0x32107654u, 0xBA98FEDCu, false, false);
    a.f = fmaxf(a.f, b.f);
    return a.f;
}

// ---------------------------------------------------------------------------
// Forward algorithm (denominator): per workgroup = 16 batches, 4 waves.
// Per step (2 barriers, no LDS-based max reduction):
//   [barrier] load alpha row chunk, local max8 + 3x v_permlane16 max -> m_r;
//             p = exp2((alpha-m)*log2e) into pbuf
//   [barrier] A-frag ds loads + 16x v_wmma_f32_16x16x4_f32 ; finalize:
//             alpha' = m + log(s)*ln2 + emit (mask-blended)
// ---------------------------------------------------------------------------
__global__ __launch_bounds__(128)
void crf_forward_scan(const float* __restrict__ logits,
                      const int*   __restrict__ mask,
                      const float* __restrict__ trans,
                      const float* __restrict__ startT,
                      const float* __restrict__ endT,
                      float*       __restrict__ den)
{
    __shared__ float alpha[BPG][LSTRIDE];
    __shared__ float pbuf [BPG][LSTRIDE];
    __shared__ float mrow [BPG];
    __shared__ float mskrow[BPG];

    const int tid  = threadIdx.x;
    const int lane = tid & 31;
    const int wv   = tid >> 5;        // wave id 0..3
    const int hh   = lane >> 4;       // half-wave: 0 or 1
    const int nl   = lane & 15;       // lane-in-half
    const int jt   = wv * 16;         // this wave's N-tile base column
    const int b0   = blockIdx.x * BPG;

    // B fragments: exp(transitions) for this wave's 16 columns, kept in VGPRs.
    // B layout (4x16 f32): lane holds (K = 4c + 2*hh + {0,1}, N = jt + nl).
    v2f Bfrag[16];
    #pragma unroll
    for (int c = 0; c < 16; ++c) {
        const int k = 4 * c + 2 * hh;
        const int n = jt + nl;
        Bfrag[c].x = __builtin_amdgcn_exp2f(trans[(k    ) * NUM_TAGS + n] * LOG2E);
        Bfrag[c].y = __builtin_amdgcn_exp2f(trans[(k + 1) * NUM_TAGS + n] * LOG2E);
    }

    const int r  = tid >> 3;         // row 0..15 this thread covers
    const int c0 = (tid & 7) * 8;    // 8-column chunk base

    // alpha0 = start_transitions + logits[:, 0, :]
    #pragma unroll
    for (int i = 0; i < 8; ++i) {
        const int j = c0 + i;
        alpha[r][j] = startT[j] +
            logits[((size_t)(b0 + r) * SEQ_LEN) * NUM_TAGS + j];
    }

    for (int t = 1; t < SEQ_LEN; ++t) {
        __syncthreads();   // prior finalize writes of alpha visible

        // row max via local max8 + permlane16 reduction over the row's 8 lanes
        float av[8];
        #pragma unroll
        for (int i = 0; i < 8; ++i) av[i] = alpha[r][c0 + i];
        float mx = av[0];
        #pragma unroll
        for (int i = 1; i < 8; ++i) mx = fmaxf(mx, av[i]);
        const float m_r = permlane_max8(mx);

        if ((tid & 7) == 0) mrow[r] = m_r;
        if (tid < BPG)
            mskrow[tid] = (float)mask[(size_t)(b0 + tid) * SEQ_LEN + t];
        #pragma unroll
        for (int i = 0; i < 8; ++i)
            pbuf[r][c0 + i] = __builtin_amdgcn_exp2f((av[i] - m_r) * LOG2E);
        if (t + 4 < SEQ_LEN) {   // stream emissions ~4 steps ahead
            __builtin_prefetch(
                &logits[((size_t)(b0 + r) * SEQ_LEN + (t + 4)) * NUM_TAGS + c0],
                0, 1);
        }
        __syncthreads();   // pbuf, mrow, mskrow visible

        // GEMM: D(16x16) = P(16x64) x expT(64x16) via 16 f32 WMMAs (K=4 each)
        v8f acc = {};
        #pragma unroll
        for (int c = 0; c < 16; ++c) {
            const int k0 = 4 * c + 2 * hh;  // A: lane holds (M=nl, K=k0,k0+1)
            const v2f a = *(const v2f*)&pbuf[nl][k0];
            acc = __builtin_amdgcn_wmma_f32_16x16x4_f32(
                false, a, false, Bfrag[c], (short)0, acc, false, false);
        }

        // finalize: alpha' = m + log(s) + emit, mask-blended
        // D layout: VGPR v holds (M = v + 8*hh, N = jt + nl).
        #pragma unroll
        for (int v = 0; v < 8; ++v) {
            const int row = v + 8 * hh;
            const int j   = jt + nl;
            const float s    = acc[v];
            const float emit =
                logits[((size_t)(b0 + row) * SEQ_LEN + t) * NUM_TAGS + j];
            const float newv = mrow[row] + __builtin_amdgcn_logf(s) * LN2 + emit;
            const float mf   = mskrow[row];
            const float oldv = alpha[row][j];
            alpha[row][j] = mf * newv + (1.0f - mf) * oldv;
        }
    }

    __syncthreads();
    // denominator: logsumexp_j(alpha + end_transitions)
    if (tid < BPG) {
        float mx = alpha[tid][0] + endT[0];
        for (int j = 1; j < NUM_TAGS; ++j)
            mx = fmaxf(mx, alpha[tid][j] + endT[j]);
        float s = 0.0f;
        for (int j = 0; j < NUM_TAGS; ++j)
            s += __builtin_amdgcn_exp2f((alpha[tid][j] + endT[j] - mx) * LOG2E);
        den[b0 + tid] = mx + __builtin_amdgcn_logf(s) * LN2;
    }
}

// ---------------------------------------------------------------------------
// Joint (numerator): path score gather, one block per batch.
// ---------------------------------------------------------------------------
__global__ __launch_bounds__(256)
void crf_joint(const float* __restrict__ logits,
               const int*   __restrict__ tags,
               const int*   __restrict__ mask,
               const float* __restrict__ trans,
               const float* __restrict__ startT,
               const float* __restrict__ endT,
               float*       __restrict__ num)
{
    __shared__ float sTrans[256];
    __shared__ float sEmit [256];
    __shared__ int   sMask [256];

    const int b   = blockIdx.x;
    const int tid = threadIdx.x;
    const size_t base = (size_t)b * SEQ_LEN;

    float transAcc = 0.0f, emitAcc = 0.0f;
    int   mAcc = 0;
    for (int t = tid; t < SEQ_LEN; t += 256) {
        const int mk = mask[base + t];
        const int tg = tags[base + t] * mk;
        mAcc += mk;
        const float mf = (float)mk;
        if (t >= 1) {
            const int mkp = mask[base + t - 1];
            const int tgp = tags[base + t - 1] * mkp;
            transAcc += trans[tgp * NUM_TAGS + tg] * mf;
        }
        if (t < SEQ_LEN - 1) {
            emitAcc += logits[(base + t) * NUM_TAGS + tg] * mf;
        }
    }
    sTrans[tid] = transAcc;
    sEmit [tid] = emitAcc;
    sMask [tid] = mAcc;
    __syncthreads();
    for (int off = 128; off > 0; off >>= 1) {
        if (tid < off) {
            sTrans[tid] += sTrans[tid + off];
            sEmit [tid] += sEmit [tid + off];
            sMask [tid] += sMask [tid + off];
        }
        __syncthreads();
    }
    if (tid == 0) {
        const int tg0 = tags[base] * mask[base];
        float score = startT[tg0] + sTrans[0] + sEmit[0];
        int last = sMask[0] - 1;
        if (last < 0) last += SEQ_LEN;            // python negative-index wrap
        const int ltag = tags[base + last] * mask[base + last];
        score += endT[ltag];
        score += logits[(base + SEQ_LEN - 1) * NUM_TAGS + ltag] *
                 (float)mask[base + SEQ_LEN - 1];
        num[b] = score;
    }
}

// ---------------------------------------------------------------------------
// Final deterministic reduction: sum_b (num[b] - den[b]) -> scalar.
// ---------------------------------------------------------------------------
__global__ __launch_bounds__(256)
void crf_reduce(const float* __restrict__ num,
                const float* __restrict__ den,
                float*       __restrict__ out)
{
    __shared__ float s[256];
    const int tid = threadIdx.x;
    s[tid] = num[tid] - den[tid];
    __syncthreads();
    for (int off = 128; off > 0; off >>= 1) {
        if (tid < off) s[tid] += s[tid + off];
        __syncthreads();
    }
    if (tid == 0) out[0] = s[0];
}

extern "C" void kernel_launch(void* const* d_in, const int* in_sizes, int n_in,
                              void* d_out, int out_size, void* d_ws, size_t ws_size,
                              hipStream_t stream)
{
    const float* logits = (const float*)d_in[0];   // (256, 2048, 64) f32
    const int*   tags   = (const int*)  d_in[1];   // (256, 2048) i32
    const int*   mask   = (const int*)  d_in[2];   // (256, 2048) i32
    const float* trans  = (const float*)d_in[3];   // (64, 64) f32
    const float* startT = (const float*)d_in[4];   // (64,) f32
    const float* endT   = (const float*)d_in[5];   // (64,) f32

    float* num = (float*)d_ws;          // [256]
    float* den = num + BSZ;             // [256]

    crf_forward_scan<<<BSZ / BPG, 128, 0, stream>>>(logits, mask, trans,
                                                    startT, endT, den);
    crf_joint<<<BSZ, 256, 0, stream>>>(logits, tags, mask, trans,
                                       startT, endT, num);
    crf_reduce<<<1, 256, 0, stream>>>(num, den, (float*)d_out);
}